// VRAttention_69956427317524
// MI455X (gfx1250) — compile-verified
//
#include <hip/hip_runtime.h>
#include <hip/hip_bf16.h>
#include <math.h>

typedef __attribute__((ext_vector_type(2))) float v2f;
typedef __attribute__((ext_vector_type(8))) float v8f;

#define BN    4
#define CINC  128
#define PLC   256
#define ATTC  64
#define RRC   16
#define HWD   4096      // 64*64
#define NSTEP 6
#define EPSV  1e-6f

// ---------------------------------------------------------------------------
// f32 WMMA micro-kernel: D(16x16) = A(16xK) * B(Kx16), K stepped by 4.
// A row-major (ldA), B row-major (ldB).  Optional per-k scale on B rows.
// Lane mapping (ISA 7.12.2):
//   A: lane<16 -> K = k0+0,k0+1 ; lane>=16 -> K = k0+2,k0+3 (rows = lane&15)
//   B: symmetric (cols = lane&15)
//   D: vgpr i -> row i + 8*(lane>>4), col = lane&15
// ---------------------------------------------------------------------------
__device__ __forceinline__ v8f wmma_k_f32(const float* __restrict__ A, int ldA, int a_row0,
                                          const float* __restrict__ Bm, int ldB, int b_col0,
                                          int K, const float* __restrict__ scale,
                                          int lq, int half)
{
  v8f acc = {};
  const float* arow = A + (size_t)(a_row0 + lq) * ldA + 2 * half;
  const float* bcol = Bm + b_col0 + lq + (size_t)(2 * half) * ldB;
#pragma unroll 4
  for (int k0 = 0; k0 < K; k0 += 4) {
    v2f a;  a.x = arow[k0];              a.y = arow[k0 + 1];
    v2f bb; bb.x = bcol[(size_t)k0 * ldB]; bb.y = bcol[(size_t)(k0 + 1) * ldB];
    if (scale) { bb.x *= scale[k0 + 2 * half]; bb.y *= scale[k0 + 2 * half + 1]; }
    acc = __builtin_amdgcn_wmma_f32_16x16x4_f32(false, a, false, bb, (short)0, acc,
                                                false, false);
  }
  return acc;
}

// ---------------------------------------------------------------------------
// 1) y[b,o,hw] = sum_c conv1_w[o,c] * x[b,c,hw]      (O=256, C=128)
// ---------------------------------------------------------------------------
__global__ void __launch_bounds__(128) k_conv1(const float* __restrict__ x,
                                               const float* __restrict__ w1,
                                               float* __restrict__ y)
{
  int wid  = blockIdx.x * 4 + (threadIdx.x >> 5);
  int lane = threadIdx.x & 31, lq = lane & 15, half = lane >> 4;
  int b   = wid >> 12;               // 16*256 = 4096 tiles per batch
  int r   = wid & 4095;
  int o0  = (r >> 8) << 4;
  int hw0 = (r & 255) << 4;
  const float* Bm = x + (size_t)b * CINC * HWD;
  v8f acc = wmma_k_f32(w1, CINC, o0, Bm, HWD, hw0, CINC, nullptr, lq, half);
  float* yp = y + ((size_t)b * PLC + o0 + 8 * half) * HWD + hw0 + lq;
#pragma unroll
  for (int i = 0; i < 8; ++i) yp[(size_t)i * HWD] = acc[i];
}

// ---------------------------------------------------------------------------
// 2) depthwise 3x3 SAME conv + bias
// ---------------------------------------------------------------------------
__global__ void __launch_bounds__(256) k_dw(const float* __restrict__ y,
                                            const float* __restrict__ w3,
                                            const float* __restrict__ b3,
                                            float* __restrict__ y2)
{
  int idx = blockIdx.x * 256 + threadIdx.x;      // 4*256*4096 total
  int w  = idx & 63, h = (idx >> 6) & 63, ch = (idx >> 12) & 255, b = idx >> 20;
  const float* src = y + ((size_t)b * PLC + ch) * HWD;
  const float* kw  = w3 + ch * 9;
  float acc = b3[ch];
#pragma unroll
  for (int dy = -1; dy <= 1; ++dy) {
    int hh = h + dy; if ((unsigned)hh >= 64u) continue;
#pragma unroll
    for (int dx = -1; dx <= 1; ++dx) {
      int ww = w + dx; if ((unsigned)ww >= 64u) continue;
      acc += src[hh * 64 + ww] * kw[(dy + 1) * 3 + (dx + 1)];
    }
  }
  y2[idx] = acc;
}

// ---------------------------------------------------------------------------
// 3) xl[m,p] = relu( sum_c y2[b,c,hw]*lower_w[p,c] + lower_b[p] ), pixel-major
// ---------------------------------------------------------------------------
__global__ void __launch_bounds__(128) k_lower(const float* __restrict__ y2,
                                               const float* __restrict__ lw,
                                               const float* __restrict__ lb,
                                               float* __restrict__ xl)
{
  int wid  = blockIdx.x * 4 + (threadIdx.x >> 5);
  int lane = threadIdx.x & 31, lq = lane & 15, half = lane >> 4;
  int b   = wid >> 12;
  int r   = wid & 4095;
  int o0  = (r >> 8) << 4;
  int hw0 = (r & 255) << 4;
  const float* Bm = y2 + (size_t)b * PLC * HWD;
  v8f acc = wmma_k_f32(lw, PLC, o0, Bm, HWD, hw0, PLC, nullptr, lq, half);
  int m0 = b * HWD + hw0 + lq;       // pixel index for this lane
  int p0 = o0 + 8 * half;
  float* xp = xl + (size_t)m0 * PLC + p0;
#pragma unroll
  for (int i = 0; i < 8; ++i) xp[i] = fmaxf(acc[i] + lb[p0 + i], 0.f);
}

// ---------------------------------------------------------------------------
// 4) per-pixel EM (D=1 => rank-1 collapse) + feat = relu(xm + xr)
//    one wave32 per pixel; lane owns rows n = lane + 32j, j=0..7
// ---------------------------------------------------------------------------
__global__ void __launch_bounds__(128) k_em(const float* __restrict__ xl,
                                            const float* __restrict__ binit,
                                            const float* __restrict__ y2,
                                            float* __restrict__ feat)
{
  int m    = blockIdx.x * 4 + (threadIdx.x >> 5);   // pixel 0..16383
  int lane = threadIdx.x & 31;
  int b = m >> 12, hw = m & 4095;

  float x[8];
#pragma unroll
  for (int j = 0; j < 8; ++j) x[j] = xl[(size_t)m * PLC + lane + 32 * j];

  float t = (lane < 16) ? binit[(size_t)m * RRC + lane] : 0.f;
  float bas[16];
#pragma unroll
  for (int r = 0; r < 16; ++r) bas[r] = __shfl(t, r, 32);
#pragma unroll
  for (int r = 0; r < 16; ++r) bas[r] = bas[r] / fmaxf(fabsf(bas[r]), 1e-12f);

  // coef = softmax_r( x[n]*bases[r] )
  float coef[8][16];
#pragma unroll
  for (int j = 0; j < 8; ++j) {
    float mx = -3.4e38f;
#pragma unroll
    for (int r = 0; r < 16; ++r) mx = fmaxf(mx, x[j] * bas[r]);
    float s = 0.f;
#pragma unroll
    for (int r = 0; r < 16; ++r) { float e = __expf(x[j] * bas[r] - mx); coef[j][r] = e; s += e; }
    float inv = 1.f / s;
#pragma unroll
    for (int r = 0; r < 16; ++r) coef[j][r] *= inv;
  }

  for (int step = 0; step < NSTEP; ++step) {
    float pn[16], pd[16];
#pragma unroll
    for (int r = 0; r < 16; ++r) { pn[r] = 0.f; pd[r] = 0.f; }
#pragma unroll
    for (int j = 0; j < 8; ++j) {
      float S = 0.f;
#pragma unroll
      for (int r = 0; r < 16; ++r) S += coef[j][r] * bas[r];
#pragma unroll
      for (int r = 0; r < 16; ++r) {
        float num = x[j] * bas[r];
        float den = S * bas[r] + EPSV;
        coef[j][r] = coef[j][r] * num / den;
      }
      float S2 = 0.f;
#pragma unroll
      for (int r = 0; r < 16; ++r) S2 += coef[j][r] * bas[r];
#pragma unroll
      for (int r = 0; r < 16; ++r) { pn[r] += x[j] * coef[j][r]; pd[r] += S2 * coef[j][r]; }
    }
#pragma unroll
    for (int off = 16; off; off >>= 1) {
#pragma unroll
      for (int r = 0; r < 16; ++r) {
        pn[r] += __shfl_xor(pn[r], off, 32);
        pd[r] += __shfl_xor(pd[r], off, 32);
      }
    }
#pragma unroll
    for (int r = 0; r < 16; ++r) bas[r] = bas[r] * pn[r] / (pd[r] + EPSV);
  }

  // final coef refresh with xn, then xr[n] = sum_r bases[r]*coef[n,r]
#pragma unroll
  for (int j = 0; j < 8; ++j) {
    float S = 0.f;
#pragma unroll
    for (int r = 0; r < 16; ++r) S += coef[j][r] * bas[r];
    float xr = 0.f;
#pragma unroll
    for (int r = 0; r < 16; ++r) {
      float c2 = coef[j][r] * (x[j] * bas[r]) / (S * bas[r] + EPSV);
      xr += bas[r] * c2;
    }
    int c = lane + 32 * j;
    size_t off = ((size_t)b * PLC + c) * HWD + hw;
    feat[off] = fmaxf(y2[off] + xr, 0.f);
  }
}

// ---------------------------------------------------------------------------
// 5) avg / max pooling over hw per (b, ch)
// ---------------------------------------------------------------------------
__global__ void __launch_bounds__(256) k_pool(const float* __restrict__ feat,
                                              float* __restrict__ avg,
                                              float* __restrict__ mx)
{
  __shared__ float ss[256], sm[256];
  int bc = blockIdx.x;
  const float* p = feat + (size_t)bc * HWD;
  float s = 0.f, m = -3.4e38f;
  for (int i = threadIdx.x; i < HWD; i += 256) { float v = p[i]; s += v; m = fmaxf(m, v); }
  ss[threadIdx.x] = s; sm[threadIdx.x] = m;
  __syncthreads();
  for (int off = 128; off; off >>= 1) {
    if ((int)threadIdx.x < off) {
      ss[threadIdx.x] += ss[threadIdx.x + off];
      sm[threadIdx.x]  = fmaxf(sm[threadIdx.x], sm[threadIdx.x + off]);
    }
    __syncthreads();
  }
  if (threadIdx.x == 0) { avg[bc] = ss[0] * (1.f / HWD); mx[bc] = sm[0]; }
}

// ---------------------------------------------------------------------------
// 6) channel attention: att = sigmoid( ca2 @ (relu(ca1@avg) + relu(ca1@mx)) )
// ---------------------------------------------------------------------------
__global__ void __launch_bounds__(256) k_att(const float* __restrict__ avg,
                                             const float* __restrict__ mx,
                                             const float* __restrict__ ca1,
                                             const float* __restrict__ ca2,
                                             float* __restrict__ att,
                                             float* __restrict__ out_att)
{
  __shared__ float sa[256], sm[256], hh[64];
  int b = blockIdx.x, t = threadIdx.x;
  sa[t] = avg[b * PLC + t];
  sm[t] = mx[b * PLC + t];
  __syncthreads();
  if (t < ATTC) {
    float s1 = 0.f, s2 = 0.f;
    for (int c = 0; c < PLC; ++c) { float w = ca1[t * PLC + c]; s1 += w * sa[c]; s2 += w * sm[c]; }
    hh[t] = fmaxf(s1, 0.f) + fmaxf(s2, 0.f);
  }
  __syncthreads();
  float s = 0.f;
#pragma unroll
  for (int a = 0; a < ATTC; ++a) s += ca2[t * ATTC + a] * hh[a];
  float v = 1.f / (1.f + __expf(-s));
  att[b * PLC + t] = v;
  out_att[b * PLC + t] = v;
}

// ---------------------------------------------------------------------------
// 7) out[b,o,hw] = sum_c fc_w[o,c]*att[b,c]*feat[b,c,hw] + fc_b[o]
// ---------------------------------------------------------------------------
__global__ void __launch_bounds__(128) k_fc(const float* __restrict__ feat,
                                            const float* __restrict__ fw,
                                            const float* __restrict__ fb,
                                            const float* __restrict__ att,
                                            float* __restrict__ out)
{
  int wid  = blockIdx.x * 4 + (threadIdx.x >> 5);
  int lane = threadIdx.x & 31, lq = lane & 15, half = lane >> 4;
  int b   = wid >> 11;               // 8*256 = 2048 tiles per batch
  int r   = wid & 2047;
  int o0  = (r >> 8) << 4;
  int hw0 = (r & 255) << 4;
  v8f acc = wmma_k_f32(fw, PLC, o0, feat + (size_t)b * PLC * HWD, HWD, hw0,
                       PLC, att + b * PLC, lq, half);
  float* op = out + ((size_t)b * CINC + o0 + 8 * half) * HWD + hw0 + lq;
#pragma unroll
  for (int i = 0; i < 8; ++i) op[(size_t)i * HWD] = acc[i] + fb[o0 + 8 * half + i];
}

// ---------------------------------------------------------------------------
extern "C" void kernel_launch(void* const* d_in, const int* in_sizes, int n_in,
                              void* d_out, int out_size, void* d_ws, size_t ws_size,
                              hipStream_t stream)
{
  const float* x     = (const float*)d_in[0];
  const float* w1    = (const float*)d_in[1];
  const float* w3    = (const float*)d_in[2];
  const float* b3    = (const float*)d_in[3];
  const float* lw    = (const float*)d_in[4];
  const float* lb    = (const float*)d_in[5];
  const float* ca1   = (const float*)d_in[6];
  const float* ca2   = (const float*)d_in[7];
  const float* fw    = (const float*)d_in[8];
  const float* fb    = (const float*)d_in[9];
  const float* binit = (const float*)d_in[10];

  float* out = (float*)d_out;
  float* ws  = (float*)d_ws;

  const size_t ACT = (size_t)BN * PLC * HWD;    // 4,194,304 floats
  float* y    = ws;
  float* y2   = ws + ACT;
  float* xl   = ws + 2 * ACT;
  float* feat = ws + 3 * ACT;
  float* avg  = ws + 4 * ACT;
  float* mxp  = avg + BN * PLC;
  float* att  = mxp + BN * PLC;

  k_conv1<<<4096, 128, 0, stream>>>(x, w1, y);
  k_dw   <<<16384, 256, 0, stream>>>(y, w3, b3, y2);
  k_lower<<<4096, 128, 0, stream>>>(y2, lw, lb, xl);
  k_em   <<<4096, 128, 0, stream>>>(xl, binit, y2, feat);
  k_pool <<<BN * PLC, 256, 0, stream>>>(feat, avg, mxp);
  k_att  <<<BN, 256, 0, stream>>>(avg, mxp, ca1, ca2, att, out + (size_t)BN * CINC * HWD);
  k_fc   <<<2048, 128, 0, stream>>>(feat, fw, fb, att, out);
}